// ContrastLoss_79843442032777
// MI455X (gfx1250) — compile-verified
//
#include <hip/hip_runtime.h>
#include <math.h>

typedef __attribute__((ext_vector_type(2))) float v2f;
typedef __attribute__((ext_vector_type(8))) float v8f;

#define BDIM 4
#define CDIM 4096
#define NTHREADS 1024
#define NWAVES 32          // 1024 / wave32
#define WAVES_PER_B 8      // 256 threads per batch row
#define ELEMS_PER_THREAD 16

#define NEG_INF (-__builtin_inff())

// merge two online-logsumexp states (m, s): result represents log(s) + m
__device__ __forceinline__ void lse_merge(float& m, float& s, float mo, float so) {
    float M = fmaxf(m, mo);
    float sn;
    if (M == NEG_INF) {
        sn = 0.0f;                       // both empty -> avoid (-inf) - (-inf) NaN
    } else {
        sn = s * expf(m - M) + so * expf(mo - M);
    }
    m = M; s = sn;
}

__global__ __launch_bounds__(NTHREADS)
void contrast_loss_kernel(const float* __restrict__ contrast,
                          const int*   __restrict__ label,
                          const float* __restrict__ aux_consin,
                          const float* __restrict__ aux_label,
                          float* __restrict__ out)
{
    __shared__ float s_mN[NWAVES], s_sN[NWAVES], s_mP[NWAVES], s_sP[NWAVES];
    __shared__ float s_aux[NWAVES];
    __shared__ float s_MN[BDIM], s_MP[BDIM];
    __shared__ float s_part[64];       // [q=0..7][j=0..7], q = b*2 + {neg,pos}
    __shared__ float s_auxp[32];       // per-wave aux partial sums
    __shared__ float s_csum[16];       // WMMA column sums (contrast)
    __shared__ float s_csum2[16];      // WMMA column sums (aux)

    const int tid   = threadIdx.x;
    const int b     = tid >> 8;        // 256 threads per batch
    const int cbase = tid & 255;
    const int wave  = tid >> 5;
    const int lane  = tid & 31;

    // ---- Pass: one sweep over global memory, online logsumexp per thread ----
    float mN = NEG_INF, sN = 0.0f;     // over label==0 of  p
    float mP = NEG_INF, sP = 0.0f;     // over label==1 of -p
    float auxs = 0.0f;

    const int base = b * CDIM;
#pragma unroll
    for (int i = 0; i < ELEMS_PER_THREAD; ++i) {
        int idx = base + cbase + (i << 8);   // coalesced, stride-256 per step
        float p  = contrast[idx];
        int   l  = label[idx];
        float a  = aux_consin[idx];
        float al = aux_label[idx];
        float d  = a - al;
        auxs = fmaf(d, d, auxs);

        float v = (l == 1) ? -p : p;
        if (l == 1) {
            float M = fmaxf(mP, v);
            sP = sP * expf(mP - M) + expf(v - M);   // expf(-inf)=0 on first hit
            mP = M;
        } else {
            float M = fmaxf(mN, v);
            sN = sN * expf(mN - M) + expf(v - M);
            mN = M;
        }
    }

    // ---- Wave-level reduction (wave32 shuffles; each wave belongs to one b) ----
#pragma unroll
    for (int off = 16; off >= 1; off >>= 1) {
        float mo, so;
        mo = __shfl_xor(mN, off, 32);
        so = __shfl_xor(sN, off, 32);
        lse_merge(mN, sN, mo, so);
        mo = __shfl_xor(mP, off, 32);
        so = __shfl_xor(sP, off, 32);
        lse_merge(mP, sP, mo, so);
        auxs += __shfl_xor(auxs, off, 32);
    }

    if (lane == 0) {
        s_mN[wave] = mN; s_sN[wave] = sN;
        s_mP[wave] = mP; s_sP[wave] = sP;
        s_aux[wave] = auxs;
    }
    __syncthreads();

    // ---- Per-batch global maxima over the 8 waves of that batch ----
    if (tid < BDIM) {
        float MN = NEG_INF, MP = NEG_INF;
#pragma unroll
        for (int j = 0; j < WAVES_PER_B; ++j) {
            MN = fmaxf(MN, s_mN[tid * WAVES_PER_B + j]);
            MP = fmaxf(MP, s_mP[tid * WAVES_PER_B + j]);
        }
        s_MN[tid] = MN; s_MP[tid] = MP;
    }
    __syncthreads();

    // ---- Rescale each wave's s to the per-batch max; stage for WMMA sum ----
    if (tid < NWAVES) {
        int wb = tid >> 3;             // batch of this wave
        int wj = tid & 7;              // wave index within batch
        float MN = s_MN[wb], MP = s_MP[wb];
        float mw, sw;
        mw = s_mN[tid]; sw = s_sN[tid];
        float pN = (mw == NEG_INF) ? 0.0f : sw * expf(mw - MN);
        mw = s_mP[tid]; sw = s_sP[tid];
        float pP = (mw == NEG_INF) ? 0.0f : sw * expf(mw - MP);
        s_part[(wb * 2 + 0) * 8 + wj] = pN;
        s_part[(wb * 2 + 1) * 8 + wj] = pP;
        s_auxp[tid] = s_aux[tid];
    }
    __syncthreads();

    // ---- Cross-wave sums via V_WMMA_F32_16X16X4_F32: D = ones(16x4) x B ----
    // Column n of B holds its 4 K-slots at {lane n, lane n+16} x {V0, V1};
    // D[m,n] = column_sum(n) for every row m. Quantity q occupies columns
    // 2q (partials j=0..3) and 2q+1 (partials j=4..7).
    if (wave == 0) {                   // wave-uniform branch: EXEC all-ones
        int n  = lane & 15;
        int hi = lane >> 4;            // 0: lanes 0-15, 1: lanes 16-31

        int q  = n >> 1;
        int jb = (n & 1) * 4 + hi * 2;
        v2f Av; Av.x = 1.0f; Av.y = 1.0f;
        v2f Bv; Bv.x = s_part[q * 8 + jb];
                Bv.y = s_part[q * 8 + jb + 1];
        v8f Cz = {};
        v8f D = __builtin_amdgcn_wmma_f32_16x16x4_f32(
                    false, Av, false, Bv, (short)0, Cz, false, false);

        // aux: columns 0..7 sum the 32 per-wave aux partials (4 per column)
        int j2 = (n & 7) * 4 + hi * 2;
        float bx = s_auxp[j2];
        float by = s_auxp[j2 + 1];
        v2f Bv2; Bv2.x = (n < 8) ? bx : 0.0f;
                 Bv2.y = (n < 8) ? by : 0.0f;
        v8f D2 = __builtin_amdgcn_wmma_f32_16x16x4_f32(
                    false, Av, false, Bv2, (short)0, Cz, false, false);

        if (lane < 16) {               // row M=0 lives in D[0] of lanes 0-15
            s_csum[lane]  = D[0];
            s_csum2[lane] = D2[0];
        }
    }
    __syncthreads();

    // ---- Final scalars ----
    if (tid == 0) {
        float lc = 0.0f;
#pragma unroll
        for (int bb = 0; bb < BDIM; ++bb) {
            float SN = s_csum[4 * bb + 0] + s_csum[4 * bb + 1];
            float SP = s_csum[4 * bb + 2] + s_csum[4 * bb + 3];
            float MN = s_MN[bb], MP = s_MP[bb];
            float lse;
            if (SN <= 0.0f || SP <= 0.0f || MN == NEG_INF || MP == NEG_INF) {
                lse = NEG_INF;         // one class empty -> logaddexp(-inf,0)=0
            } else {
                lse = MN + MP + logf(SN) + logf(SP);
            }
            float contras = (lse == NEG_INF)
                          ? 0.0f
                          : fmaxf(lse, 0.0f) + log1pf(expf(-fabsf(lse)));
            lc += contras;
        }
        out[0] = lc * (1.0f / (float)BDIM);

        float at = 0.0f;
#pragma unroll
        for (int j = 0; j < 8; ++j) at += s_csum2[j];
        out[1] = at * (1.0f / (float)(BDIM * CDIM));
    }
}

extern "C" void kernel_launch(void* const* d_in, const int* in_sizes, int n_in,
                              void* d_out, int out_size, void* d_ws, size_t ws_size,
                              hipStream_t stream) {
    const float* contrast   = (const float*)d_in[0];
    const int*   label      = (const int*)  d_in[1];
    const float* aux_consin = (const float*)d_in[2];
    const float* aux_label  = (const float*)d_in[3];
    float* out = (float*)d_out;

    hipLaunchKernelGGL(contrast_loss_kernel, dim3(1), dim3(NTHREADS), 0, stream,
                       contrast, label, aux_consin, aux_label, out);
}